// DiscriminationModule_55525337203343
// MI455X (gfx1250) — compile-verified
//
#include <hip/hip_runtime.h>
#include <math.h>
#include <stdint.h>

typedef __attribute__((ext_vector_type(2))) float v2f;
typedef __attribute__((ext_vector_type(4))) float v4f;
typedef __attribute__((ext_vector_type(8))) float v8f;

#define K_DIM 32768
#define M_DIM 1024
#define ROW_CHUNK 512
#define N_CHUNKS (K_DIM / ROW_CHUNK) /* 64 */
#define CG_ITERS 40
#define ALPHA_THR 1.0f

/* ---------------- Kernel 1a: per-column squared-norm and x-dot partials ---- */
__global__ __launch_bounds__(256)
void dm_colstats_partial(const float* __restrict__ W, const float* __restrict__ x,
                         float* __restrict__ pn, float* __restrict__ pd) {
  int col = blockIdx.y * 256 + threadIdx.x;      // 0..1023, coalesced
  int r0  = blockIdx.x * ROW_CHUNK;
  float n = 0.f, d = 0.f;
  for (int r = 0; r < ROW_CHUNK; ++r) {
    float w  = W[(size_t)(r0 + r) * M_DIM + col];
    float xv = x[r0 + r];
    n = fmaf(w, w, n);
    d = fmaf(xv, w, d);
  }
  pn[blockIdx.x * M_DIM + col] = n;
  pd[blockIdx.x * M_DIM + col] = d;
}

/* ---------------- Kernel 1b: finalize invn and rhs b ---------------------- */
__global__ __launch_bounds__(1024)
void dm_colstats_final(const float* __restrict__ pn, const float* __restrict__ pd,
                       float* __restrict__ invn, float* __restrict__ b) {
  int col = threadIdx.x;
  float n = 0.f, d = 0.f;
  for (int c = 0; c < N_CHUNKS; ++c) {   // fixed order: deterministic
    n += pn[c * M_DIM + col];
    d += pd[c * M_DIM + col];
  }
  float nrm = fmaxf(sqrtf(n), 1e-12f);   // torch F.normalize eps
  float iv  = 1.0f / nrm;
  invn[col] = iv;
  b[col]    = d * iv;                    // out_ = (x @ W_raw) * invn
}

/* ---------------- Kernel 2: R = D (W_raw^T W_raw) D via fp32 WMMA ---------
 * 128x128 tile per workgroup, 8 waves each owning a 32x64 sub-tile.
 * Global->LDS staging uses CDNA5 async copies (ASYNCcnt) with double
 * buffering so the DMA for chunk c+1 overlaps the WMMAs for chunk c.     */
#define TILE 128
#define KCH  16
#define LDSTR (TILE + 4)

__device__ __forceinline__ void dm_async_b128(uint32_t lds_off, uint64_t gaddr) {
  asm volatile("global_load_async_to_lds_b128 %0, %1, off"
               :: "v"(lds_off), "v"(gaddr) : "memory");
}
__device__ __forceinline__ void dm_wait_async0() {
  asm volatile("s_wait_asynccnt 0x0" ::: "memory");
}

__global__ __launch_bounds__(256)
void dm_syrk_wmma(const float* __restrict__ W, const float* __restrict__ invn,
                  float* __restrict__ R) {
  __shared__ float lds_a[2][KCH][LDSTR];
  __shared__ float lds_b[2][KCH][LDSTR];

  const int i0   = blockIdx.x * TILE;
  const int j0   = blockIdx.y * TILE;
  const int tid  = threadIdx.x;
  const int lane = tid & 31;
  const int wid  = tid >> 5;           // 8 waves
  const int wi   = (wid & 3) * 32;     // wave row offset in tile
  const int wj   = (wid >> 2) * 64;    // wave col offset in tile

  v8f acc[2][4] = {};

  const int m  = lane & 15;            // M/N index within 16
  const int ks = (lane >> 4) * 2;      // lanes 0-15 -> K{0,1}, 16-31 -> K{2,3}

  /* per-thread staging slots: 2 float4 per panel per chunk */
  const int row0 = tid >> 5, q0 = tid & 31;          // slot tid
  const int row1 = (tid + 256) >> 5, q1 = tid & 31;  // slot tid+256

  auto issue_chunk = [&](int k0, int buf) {
    uint64_t ga0 = (uint64_t)(uintptr_t)&W[(size_t)(k0 + row0) * M_DIM + i0 + q0 * 4];
    uint64_t gb0 = (uint64_t)(uintptr_t)&W[(size_t)(k0 + row0) * M_DIM + j0 + q0 * 4];
    uint64_t ga1 = (uint64_t)(uintptr_t)&W[(size_t)(k0 + row1) * M_DIM + i0 + q1 * 4];
    uint64_t gb1 = (uint64_t)(uintptr_t)&W[(size_t)(k0 + row1) * M_DIM + j0 + q1 * 4];
    dm_async_b128((uint32_t)(uintptr_t)&lds_a[buf][row0][q0 * 4], ga0);
    dm_async_b128((uint32_t)(uintptr_t)&lds_b[buf][row0][q0 * 4], gb0);
    dm_async_b128((uint32_t)(uintptr_t)&lds_a[buf][row1][q1 * 4], ga1);
    dm_async_b128((uint32_t)(uintptr_t)&lds_b[buf][row1][q1 * 4], gb1);
  };

  issue_chunk(0, 0);

  const int NCH = K_DIM / KCH;
  for (int c = 0; c < NCH; ++c) {
    dm_wait_async0();          // my async writes for chunk c have landed
    __syncthreads();           // everyone's writes landed; prior buf reads done
    if (c + 1 < NCH) issue_chunk((c + 1) * KCH, (c + 1) & 1);

    const float (*A)[LDSTR] = lds_a[c & 1];
    const float (*B)[LDSTR] = lds_b[c & 1];

#pragma unroll
    for (int kk = 0; kk < KCH; kk += 4) {
      v2f af[2], bf[4];
#pragma unroll
      for (int ib = 0; ib < 2; ++ib) {
        af[ib].x = A[kk + ks + 0][wi + ib * 16 + m];
        af[ib].y = A[kk + ks + 1][wi + ib * 16 + m];
      }
#pragma unroll
      for (int jb = 0; jb < 4; ++jb) {
        bf[jb].x = B[kk + ks + 0][wj + jb * 16 + m];
        bf[jb].y = B[kk + ks + 1][wj + jb * 16 + m];
      }
#pragma unroll
      for (int ib = 0; ib < 2; ++ib)
#pragma unroll
        for (int jb = 0; jb < 4; ++jb)
          acc[ib][jb] = __builtin_amdgcn_wmma_f32_16x16x4_f32(
              false, af[ib], false, bf[jb], (short)0, acc[ib][jb], false, false);
    }
  }

  /* C/D layout: VGPR v, lanes 0-15 -> (M=v, N=lane); lanes 16-31 -> (M=v+8) */
  const int n  = lane & 15;
  const int mh = (lane >> 4) * 8;
#pragma unroll
  for (int ib = 0; ib < 2; ++ib) {
#pragma unroll
    for (int jb = 0; jb < 4; ++jb) {
      int gj   = j0 + wj + jb * 16 + n;
      float sj = invn[gj];
#pragma unroll
      for (int v = 0; v < 8; ++v) {
        int gi = i0 + wi + ib * 16 + mh + v;
        R[(size_t)gi * M_DIM + gj] = acc[ib][jb][v] * invn[gi] * sj;
      }
    }
  }
}

/* ---------------- Kernel 3: y = R * p (one block per row) ----------------- */
__global__ __launch_bounds__(256)
void dm_matvec(const float* __restrict__ R, const float* __restrict__ p,
               float* __restrict__ y) {
  __shared__ float red[256];
  int row = blockIdx.x;
  int tid = threadIdx.x;
  float s = 0.f;
#pragma unroll
  for (int c = 0; c < 4; ++c) {
    int k = tid + c * 256;
    s = fmaf(R[(size_t)row * M_DIM + k], p[k], s);
  }
  red[tid] = s;
  __syncthreads();
  for (int off = 128; off > 0; off >>= 1) {
    if (tid < off) red[tid] += red[tid + off];
    __syncthreads();
  }
  if (tid == 0) y[row] = red[0];
}

/* ---------------- Kernel 4: CG init -------------------------------------- */
__global__ __launch_bounds__(1024)
void dm_cg_init(const float* __restrict__ b, float* __restrict__ x,
                float* __restrict__ r, float* __restrict__ p,
                float* __restrict__ scal) {
  __shared__ float red[1024];
  int t = threadIdx.x;
  float bv = b[t];
  x[t] = 0.f; r[t] = bv; p[t] = bv;
  red[t] = bv * bv;
  __syncthreads();
  for (int off = 512; off > 0; off >>= 1) {
    if (t < off) red[t] += red[t + off];
    __syncthreads();
  }
  if (t == 0) scal[0] = red[0];
}

/* ---------------- Kernel 5: CG scalar update ------------------------------ */
__global__ __launch_bounds__(1024)
void dm_cg_update(float* __restrict__ x, float* __restrict__ r,
                  float* __restrict__ p, const float* __restrict__ y,
                  float* __restrict__ scal) {
  __shared__ float red[1024];
  __shared__ float sAlpha, sRR, sBeta;
  int t = threadIdx.x;
  float pv = p[t], yv = y[t];
  red[t] = pv * yv;                      // p.Ap
  __syncthreads();
  for (int off = 512; off > 0; off >>= 1) {
    if (t < off) red[t] += red[t + off];
    __syncthreads();
  }
  if (t == 0) { float rr = scal[0]; sRR = rr; sAlpha = rr / red[0]; }
  __syncthreads();
  float alpha = sAlpha;
  float xv = fmaf(alpha, pv, x[t]);
  float rv = fmaf(-alpha, yv, r[t]);
  x[t] = xv; r[t] = rv;
  red[t] = rv * rv;                      // r.r new
  __syncthreads();
  for (int off = 512; off > 0; off >>= 1) {
    if (t < off) red[t] += red[t + off];
    __syncthreads();
  }
  if (t == 0) { sBeta = red[0] / sRR; scal[0] = red[0]; }
  __syncthreads();
  p[t] = fmaf(sBeta, pv, rv);
}

/* ---------------- Kernel 6: unbiased-std threshold ------------------------ */
__global__ __launch_bounds__(1024)
void dm_threshold(const float* __restrict__ xsol, float* __restrict__ out) {
  __shared__ float red[1024];
  __shared__ float sMean, sThr;
  int t = threadIdx.x;
  float v = xsol[t];
  red[t] = v;
  __syncthreads();
  for (int off = 512; off > 0; off >>= 1) {
    if (t < off) red[t] += red[t + off];
    __syncthreads();
  }
  if (t == 0) sMean = red[0] * (1.0f / 1024.0f);
  __syncthreads();
  float d = v - sMean;
  red[t] = d * d;
  __syncthreads();
  for (int off = 512; off > 0; off >>= 1) {
    if (t < off) red[t] += red[t + off];
    __syncthreads();
  }
  if (t == 0) sThr = ALPHA_THR * sqrtf(red[0] * (1.0f / 1023.0f)); // ddof=1
  __syncthreads();
  out[t] = (v > sThr) ? v : 0.0f;
}

/* ---------------- Host-side orchestration -------------------------------- */
extern "C" void kernel_launch(void* const* d_in, const int* in_sizes, int n_in,
                              void* d_out, int out_size, void* d_ws, size_t ws_size,
                              hipStream_t stream) {
  const float* x_in = (const float*)d_in[0];   // [1, 32768]
  const float* Wraw = (const float*)d_in[1];   // [32768, 1024] row-major
  float* out = (float*)d_out;                  // [1024]

  float* ws = (float*)d_ws;
  float* pn   = ws;                 // 64*1024
  float* pd   = ws + 65536;         // 64*1024
  float* invn = ws + 131072;        // 1024
  float* b    = ws + 132096;        // 1024
  float* xs   = ws + 133120;        // 1024 (solution)
  float* rr   = ws + 134144;        // 1024
  float* pp   = ws + 135168;        // 1024
  float* yy   = ws + 136192;        // 1024
  float* scal = ws + 137216;        // scalars
  float* R    = ws + 262144;        // 1024*1024 (4 MB)

  dm_colstats_partial<<<dim3(N_CHUNKS, 4), 256, 0, stream>>>(Wraw, x_in, pn, pd);
  dm_colstats_final<<<1, 1024, 0, stream>>>(pn, pd, invn, b);
  dm_syrk_wmma<<<dim3(M_DIM / TILE, M_DIM / TILE), 256, 0, stream>>>(Wraw, invn, R);
  dm_cg_init<<<1, 1024, 0, stream>>>(b, xs, rr, pp, scal);
  for (int it = 0; it < CG_ITERS; ++it) {
    dm_matvec<<<M_DIM, 256, 0, stream>>>(R, pp, yy);
    dm_cg_update<<<1, 1024, 0, stream>>>(xs, rr, pp, yy, scal);
  }
  dm_threshold<<<1, 1024, 0, stream>>>(xs, out);
}